// BidirLSTMLayer_36532991820403
// MI455X (gfx1250) — compile-verified
//
#include <hip/hip_runtime.h>

// Problem constants (from reference): T=256, B=64, I=1024, H=1024
#define Tt 256
#define Bb 64
#define Ii 1024
#define Hh 1024
#define Gg 4096   // 4*H

typedef float v2f __attribute__((ext_vector_type(2)));
typedef float v8f __attribute__((ext_vector_type(8)));

// CDNA5 fp32 WMMA: D(16x16) = A(16x4) * B(4x16) + C, fp32 throughout.
// 8-arg form: (neg_a, A, neg_b, B, c_mod, C, reuse_a, reuse_b)
__device__ __forceinline__ v8f wmma4(v2f a, v2f b, v8f c) {
    return __builtin_amdgcn_wmma_f32_16x16x4_f32(false, a, false, b, (short)0, c,
                                                 false, false);
}

__device__ __forceinline__ float sigf(float x) { return 1.0f / (1.0f + __expf(-x)); }

// ---------------------------------------------------------------------------
// Transpose the four (4096 x 1024) weight matrices into K-major (1024 x 4096)
// so the GEMMs read B with coalesced float4 loads.
// ---------------------------------------------------------------------------
__global__ __launch_bounds__(256) void transpose_w(
    const float* __restrict__ w_ih_f, const float* __restrict__ w_hh_f,
    const float* __restrict__ w_ih_b, const float* __restrict__ w_hh_b,
    float* __restrict__ wT_ih, float* __restrict__ wT_hh)
{
    __shared__ float tile[32][33];
    const int z = blockIdx.z;
    const float* src = (z == 0) ? w_ih_f : (z == 1) ? w_hh_f : (z == 2) ? w_ih_b : w_hh_b;
    float* dst = (z == 0) ? wT_ih
               : (z == 1) ? wT_hh
               : (z == 2) ? (wT_ih + (size_t)Ii * Gg)
                          : (wT_hh + (size_t)Hh * Gg);
    const int tx = threadIdx.x, ty = threadIdx.y;
    const int gx  = blockIdx.x * 32 + tx;   // source col (i/k), 0..1023
    const int gy0 = blockIdx.y * 32;        // source row (g),   0..4095
#pragma unroll
    for (int j = 0; j < 32; j += 8)
        tile[ty + j][tx] = src[(size_t)(gy0 + ty + j) * 1024 + gx];
    __syncthreads();
    const int orow = blockIdx.x * 32;       // dest row (k)
    const int ocol = gy0 + tx;              // dest col (g)
#pragma unroll
    for (int j = 0; j < 32; j += 8)
        dst[(size_t)(orow + ty + j) * Gg + ocol] = tile[tx][ty + j];
}

__global__ __launch_bounds__(256) void zero_f(float* __restrict__ p, int n)
{
    int i = blockIdx.x * 256 + threadIdx.x;
    if (i < n) p[i] = 0.0f;
}

// ---------------------------------------------------------------------------
// xg[d][t][b][g] = sum_i x[t,b,i] * w_ih[d][g,i] + b_ih[d][g] + b_hh[d][g]
// GEMM: M = T*B = 16384, N = 4096, K = 1024. Block tile 128x128, KC=16.
// 8 wave32s: 4(M) x 2(N) grid, each wave 32x64 = 2x4 WMMA tiles.
// ---------------------------------------------------------------------------
#define BM1 128
#define BN1 128
#define KC1 16
#define LDA1 18    // 16 + 2: keeps frag reads conflict-free & 8B aligned
#define LDB1 136   // 128 + 8: K-row pairs land in disjoint bank halves

__global__ __launch_bounds__(256) void xg_gemm(
    const float* __restrict__ x, const float* __restrict__ wT_ih,
    const float* __restrict__ b_ih_f, const float* __restrict__ b_hh_f,
    const float* __restrict__ b_ih_b, const float* __restrict__ b_hh_b,
    float* __restrict__ xg)
{
    __shared__ float As[BM1 * LDA1];
    __shared__ float Bs[KC1 * LDB1];

    const int d = blockIdx.z;
    const float* wT  = wT_ih + (size_t)d * Ii * Gg;
    const float* bi  = d ? b_ih_b : b_ih_f;
    const float* bh  = d ? b_hh_b : b_hh_f;
    float* outp      = xg + (size_t)d * Tt * Bb * Gg;

    const int m0 = blockIdx.x * BM1;
    const int n0 = blockIdx.y * BN1;
    const int tid  = threadIdx.x;
    const int lane = tid & 31;
    const int wave = tid >> 5;
    const int wm = wave & 3;    // wave M offset = wm*32
    const int wn = wave >> 2;   // wave N offset = wn*64
    const int lr = lane & 15;   // M (for A) / N (for B/D) within tile
    const int lh = lane >> 4;   // K-half (A/B), M-half (D)

    v8f acc[2][4] = {};

    for (int k0 = 0; k0 < Ii; k0 += KC1) {
        {   // stage A (128x16): thread -> 8 consecutive floats of one row
            int row = tid >> 1;
            int col = (tid & 1) * 8;
            const float4* s4 =
                reinterpret_cast<const float4*>(x + (size_t)(m0 + row) * Ii + k0 + col);
            float4 a0 = s4[0];
            float4 a1 = s4[1];
            float2* d2 = reinterpret_cast<float2*>(&As[row * LDA1 + col]);
            d2[0] = make_float2(a0.x, a0.y);
            d2[1] = make_float2(a0.z, a0.w);
            d2[2] = make_float2(a1.x, a1.y);
            d2[3] = make_float2(a1.z, a1.w);
        }
        {   // stage B (16x128): coalesced from transposed weights
            int kr  = tid >> 4;
            int seg = (tid & 15) * 8;
            const float4* s4 =
                reinterpret_cast<const float4*>(wT + (size_t)(k0 + kr) * Gg + n0 + seg);
            float4 b0 = s4[0];
            float4 b1 = s4[1];
            float4* d4 = reinterpret_cast<float4*>(&Bs[kr * LDB1 + seg]);
            d4[0] = b0;
            d4[1] = b1;
        }
        __syncthreads();

#pragma unroll
        for (int kk = 0; kk < KC1; kk += 4) {
            v2f af[2], bf[4];
#pragma unroll
            for (int mi = 0; mi < 2; ++mi) {
                const float* ap = &As[(wm * 32 + mi * 16 + lr) * LDA1 + kk + lh * 2];
                af[mi].x = ap[0];
                af[mi].y = ap[1];
            }
#pragma unroll
            for (int ni = 0; ni < 4; ++ni) {
                const float* bp = &Bs[(kk + lh * 2) * LDB1 + wn * 64 + ni * 16 + lr];
                bf[ni].x = bp[0];
                bf[ni].y = bp[LDB1];
            }
#pragma unroll
            for (int mi = 0; mi < 2; ++mi)
#pragma unroll
                for (int ni = 0; ni < 4; ++ni)
                    acc[mi][ni] = wmma4(af[mi], bf[ni], acc[mi][ni]);
        }
        __syncthreads();
    }

    // Epilogue: D element [M][N] lives at reg = M%8, lane = (M/8)*16 + N
#pragma unroll
    for (int ni = 0; ni < 4; ++ni) {
        int gn = n0 + wn * 64 + ni * 16 + lr;
        float bias = bi[gn] + bh[gn];
#pragma unroll
        for (int mi = 0; mi < 2; ++mi) {
            int gm = m0 + wm * 32 + mi * 16 + lh * 8;
#pragma unroll
            for (int r = 0; r < 8; ++r)
                outp[(size_t)(gm + r) * Gg + gn] = acc[mi][ni][r] + bias;
        }
    }
}

// ---------------------------------------------------------------------------
// Per-step recurrent GEMM: gates[d] = xg[d][t_d] + h[d] @ w_hh[d]^T
// M = 64, N = 4096, K = 1024. Block tile 64x64 (64 blocks x 2 dirs = 128 WGs
// per step for occupancy). 8 waves: 2(M) x 4(N), wave tile 32x16.
// ---------------------------------------------------------------------------
#define BN2 64
#define KC2 16
#define LDA2 18
#define LDB2 72    // 64 + 8

__global__ __launch_bounds__(256) void gate_gemm(
    const float* __restrict__ h, const float* __restrict__ wT_hh,
    const float* __restrict__ xg, float* __restrict__ gates, int s)
{
    __shared__ float As[64 * LDA2];
    __shared__ float Bs[KC2 * LDB2];

    const int d   = blockIdx.z;
    const int t_d = d ? (Tt - 1 - s) : s;
    const float* hp   = h + (size_t)d * Bb * Hh;
    const float* wT   = wT_hh + (size_t)d * Hh * Gg;
    const float* xrow = xg + ((size_t)d * Tt + t_d) * Bb * Gg;
    float* gout       = gates + (size_t)d * Bb * Gg;

    const int n0 = blockIdx.x * BN2;
    const int tid  = threadIdx.x;
    const int lane = tid & 31;
    const int wave = tid >> 5;
    const int wm = wave & 1;    // M offset = wm*32
    const int wn = wave >> 1;   // N offset = wn*16
    const int lr = lane & 15;
    const int lh = lane >> 4;

    v8f acc[2] = {};

    for (int k0 = 0; k0 < Hh; k0 += KC2) {
        {   // stage A = h tile (64x16)
            int row = tid >> 2;
            int col = (tid & 3) * 4;
            float4 a0 = *reinterpret_cast<const float4*>(hp + (size_t)row * Hh + k0 + col);
            float2* d2 = reinterpret_cast<float2*>(&As[row * LDA2 + col]);
            d2[0] = make_float2(a0.x, a0.y);
            d2[1] = make_float2(a0.z, a0.w);
        }
        {   // stage B (16x64)
            int kr  = tid >> 4;
            int seg = (tid & 15) * 4;
            float4 b0 = *reinterpret_cast<const float4*>(wT + (size_t)(k0 + kr) * Gg + n0 + seg);
            *reinterpret_cast<float4*>(&Bs[kr * LDB2 + seg]) = b0;
        }
        __syncthreads();

#pragma unroll
        for (int kk = 0; kk < KC2; kk += 4) {
            v2f af[2], bf;
#pragma unroll
            for (int mi = 0; mi < 2; ++mi) {
                const float* ap = &As[(wm * 32 + mi * 16 + lr) * LDA2 + kk + lh * 2];
                af[mi].x = ap[0];
                af[mi].y = ap[1];
            }
            const float* bp = &Bs[(kk + lh * 2) * LDB2 + wn * 16 + lr];
            bf.x = bp[0];
            bf.y = bp[LDB2];
#pragma unroll
            for (int mi = 0; mi < 2; ++mi)
                acc[mi] = wmma4(af[mi], bf, acc[mi]);
        }
        __syncthreads();
    }

    const int gn = n0 + wn * 16 + lr;
#pragma unroll
    for (int mi = 0; mi < 2; ++mi) {
        int bm = wm * 32 + mi * 16 + lh * 8;
#pragma unroll
        for (int r = 0; r < 8; ++r) {
            size_t off = (size_t)(bm + r) * Gg + gn;
            gout[off] = acc[mi][r] + xrow[off];
        }
    }
}

// ---------------------------------------------------------------------------
// Pointwise LSTM cell update + output scatter. 2*B*H lanes of work.
// ---------------------------------------------------------------------------
__global__ __launch_bounds__(256) void lstm_update(
    const float* __restrict__ gates, float* __restrict__ cst, float* __restrict__ hst,
    float* __restrict__ out, float* __restrict__ hout, float* __restrict__ cout, int s)
{
    int idx = blockIdx.x * 256 + threadIdx.x;        // [d][b][n] flat
    if (idx >= 2 * Bb * Hh) return;
    int d   = idx / (Bb * Hh);
    int rem = idx - d * (Bb * Hh);
    int b   = rem / Hh;
    int n   = rem - b * Hh;

    const float* g = gates + (size_t)d * Bb * Gg + (size_t)b * Gg;
    float ig = sigf(g[n]);
    float fg = sigf(g[n + Hh]);
    float gg = tanhf(g[n + 2 * Hh]);
    float og = sigf(g[n + 3 * Hh]);

    float c_old = cst[idx];
    float c_new = fg * c_old + ig * gg;
    float h_new = og * tanhf(c_new);
    cst[idx] = c_new;
    hst[idx] = h_new;

    int t_d = d ? (Tt - 1 - s) : s;
    out[((size_t)t_d * Bb + b) * (size_t)(2 * Hh) + (size_t)d * Hh + n] = h_new;
    if (s == Tt - 1) {           // final states for both scans land at s = T-1
        hout[idx] = h_new;
        cout[idx] = c_new;
    }
}

// ---------------------------------------------------------------------------
extern "C" void kernel_launch(void* const* d_in, const int* in_sizes, int n_in,
                              void* d_out, int out_size, void* d_ws, size_t ws_size,
                              hipStream_t stream)
{
    (void)in_sizes; (void)n_in; (void)out_size; (void)ws_size;

    const float* x      = (const float*)d_in[0];
    const float* w_ih_f = (const float*)d_in[1];
    const float* w_hh_f = (const float*)d_in[2];
    const float* b_ih_f = (const float*)d_in[3];
    const float* b_hh_f = (const float*)d_in[4];
    const float* w_ih_b = (const float*)d_in[5];
    const float* w_hh_b = (const float*)d_in[6];
    const float* b_ih_b = (const float*)d_in[7];
    const float* b_hh_b = (const float*)d_in[8];

    // Workspace layout (floats). Total ~603 MB.
    float* ws    = (float*)d_ws;
    float* wT_ih = ws;                                   // 2 * I * G
    float* wT_hh = wT_ih + (size_t)2 * Ii * Gg;          // 2 * H * G
    float* xg    = wT_hh + (size_t)2 * Hh * Gg;          // 2 * T * B * G
    float* hst   = xg    + (size_t)2 * Tt * Bb * Gg;     // 2 * B * H
    float* cst   = hst   + (size_t)2 * Bb * Hh;          // 2 * B * H
    float* gates = cst   + (size_t)2 * Bb * Hh;          // 2 * B * G

    float* out  = (float*)d_out;                         // (T, B, 2H)
    float* hout = out  + (size_t)Tt * Bb * 2 * Hh;       // (2, B, H)
    float* cout = hout + (size_t)2 * Bb * Hh;            // (2, B, H)

    // 1) K-major weight copies (one shot, coalesced GEMM loads afterwards)
    transpose_w<<<dim3(Ii / 32, Gg / 32, 4), dim3(32, 8), 0, stream>>>(
        w_ih_f, w_hh_f, w_ih_b, w_hh_b, wT_ih, wT_hh);

    // 2) zero initial h, c (must be re-done every call for replay determinism)
    zero_f<<<dim3((4 * Bb * Hh + 255) / 256), dim3(256), 0, stream>>>(hst, 4 * Bb * Hh);

    // 3) big parallel input projection for both directions
    xg_gemm<<<dim3((Tt * Bb) / BM1, Gg / BN1, 2), dim3(256), 0, stream>>>(
        x, wT_ih, b_ih_f, b_hh_f, b_ih_b, b_hh_b, xg);

    // 4) sequential scan: fwd processes t=s, bwd processes t=T-1-s in lockstep
    for (int s = 0; s < Tt; ++s) {
        gate_gemm<<<dim3(Gg / BN2, 1, 2), dim3(256), 0, stream>>>(hst, wT_hh, xg, gates, s);
        lstm_update<<<dim3((2 * Bb * Hh) / 256), dim3(256), 0, stream>>>(
            gates, cst, hst, out, hout, cout, s);
    }
}